// LIFNeuronLayer_11493332484484
// MI455X (gfx1250) — compile-verified
//
#include <hip/hip_runtime.h>

// LIF scan: v = tau*v + (1-tau)*x; s = (v > vth); v -= s*vth (reset=0).
// B=64, T=256, U=1024, fp32. Memory-bound streaming scan over T.
// CDNA5 path: deep async global->LDS staging pipeline (ASYNCcnt), wave-private
// ring (no barriers), plus global_prefetch far-ahead for the cold HBM pass.

#define LIF_TAU 0.25f
#define LIF_VTH 1.0f

namespace {
constexpr int B = 64;
constexpr int T = 256;
constexpr int U = 1024;           // floats per (b, t) row
constexpr int COLS = U / 4;       // 256 float4 columns per row
constexpr int THREADS = 128;      // float4 columns handled per block
constexpr int BLK_PER_B = COLS / THREADS;   // 2 blocks cover one batch row
constexpr int DEPTH = 32;         // LDS ring slots (64 KB of the 320 KB WGP LDS)
constexpr int INFLIGHT = 24;      // max outstanding async loads per wave
constexpr unsigned TSTRIDE = (unsigned)U * 4u;       // 4096 bytes per timestep
constexpr unsigned LSTRIDE = (unsigned)THREADS * 16u; // 2048 bytes per ring slot
}

__device__ __forceinline__ float lif_step(float& v, float x) {
  // Bitwise-identical to reference: 0.25f*v is exact (pow2), 0.75f*x rounds
  // once, FMA single-rounds the add just like the reference's separate add.
  v = fmaf(LIF_TAU, v, (1.0f - LIF_TAU) * x);
  float s = (v > LIF_VTH) ? 1.0f : 0.0f;
  v -= s;                        // vth=1, reset=0
  return s;
}

__global__ __launch_bounds__(THREADS) void lif_async_kernel(
    const float* __restrict__ in, float* __restrict__ out) {
  __shared__ float4 ring[DEPTH][THREADS];

  const int tid = threadIdx.x;
  const int b   = blockIdx.x >> 1;                       // batch row
  const int col = ((blockIdx.x & 1) << 7) + tid;         // float4 column 0..255

  // Byte offset of (b, t=0, col) in the input tensor.
  const unsigned gbase = ((unsigned)(b * T) * (unsigned)COLS + (unsigned)col) * 16u;
  // LDS byte offset of this lane's column in ring slot 0 (low 32 bits of the
  // generic pointer to a __shared__ object are the LDS allocation offset).
  const unsigned lbase = (unsigned)(size_t)(&ring[0][0]) + ((unsigned)tid << 4);
  const unsigned long long gsaddr = (unsigned long long)(size_t)in;

  // ---- Prologue: stage timesteps 0..INFLIGHT-1 into ring slots 0..23 ----
  #pragma unroll
  for (int t = 0; t < INFLIGHT; ++t) {
    unsigned goff = gbase + (unsigned)t * TSTRIDE;
    unsigned loff = lbase + (unsigned)t * LSTRIDE;
    asm volatile("global_load_async_to_lds_b128 %0, %1, %2"
                 :: "v"(loff), "v"(goff), "s"(gsaddr) : "memory");
  }

  float4* __restrict__ out4 = reinterpret_cast<float4*>(out);
  const unsigned obase = (unsigned)(b * T) * (unsigned)COLS + (unsigned)col;

  float v0 = 0.0f, v1 = 0.0f, v2 = 0.0f, v3 = 0.0f;

  // ---- Main loop: issue t+INFLIGHT, wait for slot t, compute, store ----
  #pragma unroll 4
  for (int t = 0; t < T - INFLIGHT; ++t) {
    // Far-ahead prefetch into cache hierarchy for the cold HBM pass.
    if (t + 96 < T) {
      __builtin_prefetch(in + (size_t)(b * T + t + 96) * U + (size_t)col * 4);
    }
    // Stage timestep t+24 into slot (t+24)&31: its previous occupant (t-8)
    // was consumed 8 iterations ago, so no LDS WAR hazard.
    {
      unsigned goff = gbase + (unsigned)(t + INFLIGHT) * TSTRIDE;
      unsigned loff = lbase + (unsigned)((t + INFLIGHT) & (DEPTH - 1)) * LSTRIDE;
      asm volatile("global_load_async_to_lds_b128 %0, %1, %2"
                   :: "v"(loff), "v"(goff), "s"(gsaddr) : "memory");
    }
    // issued = INFLIGHT+t+1 total; cnt<=23 => oldest t+2 done => slot t ready
    asm volatile("s_wait_asynccnt 23" ::: "memory");

    float4 x = ring[t & (DEPTH - 1)][tid];
    float s0 = lif_step(v0, x.x);
    float s1 = lif_step(v1, x.y);
    float s2 = lif_step(v2, x.z);
    float s3 = lif_step(v3, x.w);
    out4[obase + (unsigned)t * (unsigned)COLS] = make_float4(s0, s1, s2, s3);
  }

  // ---- Drain: everything staged; finish last 24 timesteps ----
  asm volatile("s_wait_asynccnt 0" ::: "memory");

  #pragma unroll 4
  for (int t = T - INFLIGHT; t < T; ++t) {
    float4 x = ring[t & (DEPTH - 1)][tid];
    float s0 = lif_step(v0, x.x);
    float s1 = lif_step(v1, x.y);
    float s2 = lif_step(v2, x.z);
    float s3 = lif_step(v3, x.w);
    out4[obase + (unsigned)t * (unsigned)COLS] = make_float4(s0, s1, s2, s3);
  }
}

extern "C" void kernel_launch(void* const* d_in, const int* in_sizes, int n_in,
                              void* d_out, int out_size, void* d_ws, size_t ws_size,
                              hipStream_t stream) {
  (void)in_sizes; (void)n_in; (void)out_size; (void)d_ws; (void)ws_size;
  const float* in = (const float*)d_in[0];
  float* out = (float*)d_out;
  lif_async_kernel<<<dim3(B * BLK_PER_B), dim3(THREADS), 0, stream>>>(in, out);
}